// ConvMemoryMultiHead_4080218931757
// MI455X (gfx1250) — compile-verified
//
#include <hip/hip_runtime.h>

typedef __attribute__((ext_vector_type(2))) float v2f;
typedef __attribute__((ext_vector_type(8))) float v8f;

#define IN_CH   128
#define KEY_CH  256   // N_HEADS*HEAD_DIM
#define HW      4096  // 64*64
#define IMG_W   64
#define NHEADS  8
#define CDIM    288   // HEAD_DIM*3*3
#define HEAD_OUT 32   // OUT_CH / N_HEADS
#define LDS_STRIDE 290  // 288 padded to break bank conflicts, kept even for b64

// ---------------------------------------------------------------------------
// Kernel 1: key_map[b][o][l] = sum_i w_key[o][i] * x[b][i][l] + b_key[o]
// GEMM M=256, K=128, N=4096 per batch via V_WMMA_F32_16X16X4_F32.
// Block = 8 waves: 4 M-groups (64 rows each) x 2 N-groups (16 cols each).
// ---------------------------------------------------------------------------
__global__ __launch_bounds__(256) void keymap_gemm(
    const float* __restrict__ x, const float* __restrict__ w_key,
    const float* __restrict__ b_key, float* __restrict__ key_map)
{
    const int lane = threadIdx.x & 31;
    const int wave = threadIdx.x >> 5;
    const int lo16 = lane & 15;
    const int hi   = lane >> 4;                 // 0 or 1 (half-wave)
    const int m0   = (wave & 3) * 64;
    const int n0   = blockIdx.x * 32 + (wave >> 2) * 16;
    const int b    = blockIdx.z;
    const float* xb = x + (size_t)b * IN_CH * HW;

    v8f acc[4] = {};
    for (int k0 = 0; k0 < IN_CH; k0 += 4) {
        const int kb = k0 + 2 * hi;             // lane's K base (K = kb + v)
        // B fragment: B[K][N] = x[b][kb+v][n0+lo16]
        v2f bfrag;
        bfrag.x = xb[(size_t)kb * HW + n0 + lo16];
        bfrag.y = xb[(size_t)(kb + 1) * HW + n0 + lo16];
#pragma unroll
        for (int st = 0; st < 4; ++st) {
            const int row = m0 + st * 16 + lo16;   // A row (M = lane%16)
            v2f afrag = *(const v2f*)(w_key + (size_t)row * IN_CH + kb);
            acc[st] = __builtin_amdgcn_wmma_f32_16x16x4_f32(
                false, afrag, false, bfrag, (short)0, acc[st], false, false);
        }
    }

    float* km = key_map + (size_t)b * KEY_CH * HW;
#pragma unroll
    for (int st = 0; st < 4; ++st) {
#pragma unroll
        for (int r = 0; r < 8; ++r) {
            const int m = m0 + st * 16 + r + 8 * hi;   // D: VGPR r -> M = r + 8*hi
            km[(size_t)m * HW + n0 + lo16] = acc[st][r] + b_key[m];
        }
    }
}

// ---------------------------------------------------------------------------
// Kernel 2: one block (1 wave) per (batch, head, 16-wide spatial strip).
// Stage 1: gather raw attn scores [16 pos][288 c] into LDS
// Stage 2: softmax over c per position (split across half-waves + shfl_xor)
// Stage 3: out(32x16) = memT(32x288) @ attn(288x16) with fp32 WMMA
// ---------------------------------------------------------------------------
__global__ __launch_bounds__(32) void attn_mem_gemm(
    const float* __restrict__ key_map, const float* __restrict__ memory,
    float* __restrict__ out)
{
    __shared__ float att[16 * LDS_STRIDE];

    const int lane = threadIdx.x;
    const int lo16 = lane & 15;
    const int hi   = lane >> 4;
    const int l0   = blockIdx.x * 16;       // strip start (within one image row)
    const int n    = blockIdx.y;            // head
    const int b    = blockIdx.z;
    const int y    = l0 >> 6;               // l0 / 64
    const int x0   = l0 & 63;
    const float* km = key_map + (size_t)b * KEY_CH * HW;

    // ---- Stage 1: raw scores. attn[c] for head n uses unfold index g = c*8+n.
    for (int idx = lane; idx < 16 * CDIM; idx += 32) {
        const int j  = idx / CDIM;          // position within strip
        const int c  = idx - j * CDIM;
        const int g  = c * NHEADS + n;
        const int ch = g / 9;
        const int kk = g - ch * 9;
        const int kh = kk / 3;
        const int kw = kk - kh * 3;
        const int yy = y + kh - 1;
        const int xx = x0 + j + kw - 1;
        float v = 0.0f;
        if ((unsigned)yy < 64u && (unsigned)xx < 64u)
            v = km[((size_t)ch * IMG_W + yy) * IMG_W + xx];
        att[j * LDS_STRIDE + c] = v;
    }
    __syncthreads();

    // ---- Stage 2: softmax over c (288) per position. Lane (lo16,hi) owns half
    // the c-range of row lo16; combine halves with a cross-16 shuffle.
    {
        float* row = att + lo16 * LDS_STRIDE + hi * 144;
        float mx = -3.402823466e38f;
        for (int c = 0; c < 144; ++c) mx = fmaxf(mx, row[c]);
        mx = fmaxf(mx, __shfl_xor(mx, 16, 32));
        float s = 0.0f;
        for (int c = 0; c < 144; ++c) {
            const float e = __expf(row[c] - mx);
            row[c] = e;
            s += e;
        }
        s += __shfl_xor(s, 16, 32);
        const float inv = 1.0f / s;
        for (int c = 0; c < 144; ++c) row[c] *= inv;
    }
    __syncthreads();

    // ---- Stage 3: GEMM. A[i][c] = memory[n][c][i] (32x288), B = attn (288x16).
    const float* mem = memory + (size_t)n * CDIM * HEAD_OUT;
    v8f acc0 = {}, acc1 = {};
    for (int k0 = 0; k0 < CDIM; k0 += 4) {
        const int kb = k0 + 2 * hi;
        const v2f bfrag = *(const v2f*)(att + lo16 * LDS_STRIDE + kb); // ds_load_b64
        v2f a0, a1;
        a0.x = mem[(size_t)kb * HEAD_OUT + lo16];
        a0.y = mem[(size_t)(kb + 1) * HEAD_OUT + lo16];
        a1.x = mem[(size_t)kb * HEAD_OUT + 16 + lo16];
        a1.y = mem[(size_t)(kb + 1) * HEAD_OUT + 16 + lo16];
        acc0 = __builtin_amdgcn_wmma_f32_16x16x4_f32(
            false, a0, false, bfrag, (short)0, acc0, false, false);
        acc1 = __builtin_amdgcn_wmma_f32_16x16x4_f32(
            false, a1, false, bfrag, (short)0, acc1, false, false);
    }

    // out channel = n*32 + i, flat layout [8][256][4096]
    float* ob = out + (size_t)(b * NHEADS + n) * HEAD_OUT * HW + l0;
#pragma unroll
    for (int r = 0; r < 8; ++r) {
        ob[(size_t)(r + 8 * hi) * HW + lo16]      = acc0[r];
        ob[(size_t)(16 + r + 8 * hi) * HW + lo16] = acc1[r];
    }
}

// ---------------------------------------------------------------------------
extern "C" void kernel_launch(void* const* d_in, const int* in_sizes, int n_in,
                              void* d_out, int out_size, void* d_ws, size_t ws_size,
                              hipStream_t stream) {
    (void)in_sizes; (void)n_in; (void)out_size; (void)ws_size;
    const float* x      = (const float*)d_in[0];   // [8,128,64,64]
    const float* w_key  = (const float*)d_in[1];   // [256,128]
    const float* b_key  = (const float*)d_in[2];   // [256]
    const float* memory = (const float*)d_in[3];   // [8,288,32]
    float* out = (float*)d_out;                    // [8,256,64,64]
    float* key_map = (float*)d_ws;                 // [8,256,4096] = 33.5 MB scratch

    // Kernel 1: conv-as-GEMM. grid: 128 N-tiles (32 cols each) x 8 batches.
    keymap_gemm<<<dim3(128, 1, 8), 256, 0, stream>>>(x, w_key, b_key, key_map);

    // Kernel 2: gather+softmax+memory GEMM. grid: 256 strips x 8 heads x 8 batches.
    attn_mem_gemm<<<dim3(256, 8, 8), 32, 0, stream>>>(key_map, memory, out);
}